// BLR_31009663877255
// MI455X (gfx1250) — compile-verified
//
#include <hip/hip_runtime.h>
#include <math.h>

// Problem constants (match reference)
#define BN 32
#define SN 512
#define QN 512
#define ID 256   // IN_DIM
#define OD 64    // OUT_DIM
#define SIG_EPS 0.1f
#define STR 260  // padded LDS row stride (words); 260 % 64 == 4 -> bank-friendly

typedef float v2f __attribute__((ext_vector_type(2)));
typedef float v4f __attribute__((ext_vector_type(4)));
typedef float v8f __attribute__((ext_vector_type(8)));

// D = A(16x4,f32) * B(4x16,f32) + C(16x16,f32), full f32 precision.
__device__ __forceinline__ v8f wmma_k4(v2f a, v2f b, v8f c) {
  return __builtin_amdgcn_wmma_f32_16x16x4_f32(false, a, false, b, (short)0, c,
                                               false, false);
}

// ---------------- small setup kernels ----------------

__global__ void k_init(float* __restrict__ spread, float* __restrict__ nll) {
  int t = blockIdx.x * blockDim.x + threadIdx.x;
  if (t < BN * QN) spread[t] = 1.0f;
  if (t == 0) *nll = 0.0f;
}

// L = L_asym @ L_asym^T   (L[i][j] = dot(row i, row j))
__global__ void k_L(const float* __restrict__ La, float* __restrict__ L) {
  int t = blockIdx.x * blockDim.x + threadIdx.x;
  if (t >= ID * ID) return;
  int i = t / ID, j = t % ID;
  const float* ri = La + (size_t)i * ID;
  const float* rj = La + (size_t)j * ID;
  float s = 0.0f;
  for (int k = 0; k < ID; ++k) s = fmaf(ri[k], rj[k], s);
  L[t] = s;
}

// LK = L @ K
__global__ void k_LK(const float* __restrict__ L, const float* __restrict__ K,
                     float* __restrict__ LK) {
  int t = blockIdx.x * blockDim.x + threadIdx.x;
  if (t >= ID * OD) return;
  int i = t / OD, o = t % OD;
  float s = 0.0f;
  for (int j = 0; j < ID; ++j) s = fmaf(L[(size_t)i * ID + j], K[(size_t)j * OD + o], s);
  LK[t] = s;
}

// ---------------- WMMA GEMM kernels (one wave = one 16x16 C tile) ----------

// posterior_L[b] = PhiS[b]^T @ PhiS[b] + L   -> PL [B,256,256]
__global__ void k_postL(const float* __restrict__ phiS, const float* __restrict__ L,
                        float* __restrict__ PL) {
  int lane = threadIdx.x & 31;
  int tile = blockIdx.x * 4 + (threadIdx.x >> 5);  // 0..255
  int b = blockIdx.y;
  int it = (tile >> 4) << 4;
  int jt = (tile & 15) << 4;
  const float* Phi = phiS + (size_t)b * SN * ID;
  int m = lane & 15;
  int bk = (lane >> 4) << 1;          // 0 or 2
  v8f c = {};
  for (int s0 = 0; s0 < SN; s0 += 4) {
    v2f a, bb;
    a.x = Phi[(size_t)(s0 + bk + 0) * ID + it + m];  // A(m,k)=Phi[s][it+m]
    a.y = Phi[(size_t)(s0 + bk + 1) * ID + it + m];
    bb.x = Phi[(size_t)(s0 + bk + 0) * ID + jt + m]; // B(k,n)=Phi[s][jt+n]
    bb.y = Phi[(size_t)(s0 + bk + 1) * ID + jt + m];
    c = wmma_k4(a, bb, c);
  }
  float* out = PL + (size_t)b * ID * ID;
  int rb = (lane >> 4) << 3;
  #pragma unroll
  for (int v = 0; v < 8; ++v) {
    int row = it + rb + v;
    out[(size_t)row * ID + jt + m] = c[v] + L[(size_t)row * ID + jt + m];
  }
}

// RHS[b] = PhiS[b]^T @ Y[b] + LK   -> [B,256,64]
__global__ void k_rhs(const float* __restrict__ phiS, const float* __restrict__ Y,
                      const float* __restrict__ LK, float* __restrict__ RHS) {
  int lane = threadIdx.x & 31;
  int tile = blockIdx.x * 4 + (threadIdx.x >> 5);  // 0..63
  int b = blockIdx.y;
  int it = (tile >> 2) << 4;
  int jt = (tile & 3) << 4;
  const float* Phi = phiS + (size_t)b * SN * ID;
  const float* Yb = Y + (size_t)b * SN * OD;
  int m = lane & 15;
  int bk = (lane >> 4) << 1;
  v8f c = {};
  for (int s0 = 0; s0 < SN; s0 += 4) {
    v2f a, bb;
    a.x = Phi[(size_t)(s0 + bk + 0) * ID + it + m];
    a.y = Phi[(size_t)(s0 + bk + 1) * ID + it + m];
    bb.x = Yb[(size_t)(s0 + bk + 0) * OD + jt + m];
    bb.y = Yb[(size_t)(s0 + bk + 1) * OD + jt + m];
    c = wmma_k4(a, bb, c);
  }
  float* out = RHS + (size_t)b * ID * OD;
  int rb = (lane >> 4) << 3;
  #pragma unroll
  for (int v = 0; v < 8; ++v) {
    int row = it + rb + v;
    out[(size_t)row * OD + jt + m] = c[v] + LK[(size_t)row * OD + jt + m];
  }
}

// ---------------- blocked Gauss-Jordan inverse, WMMA trailing update -------
// One workgroup (256 thr) per batch; full 256x256 matrix in LDS (266 KB).
// Per 16-wide pivot block K:
//   A: pivot block -> P^-1 in place (elementwise, 2 syncs/step)
//   B: A[K,J] <- P^-1 * A[K,J]                (WMMA, wave-parallel over J)
//   C: A[I,J] <- A[I,J] - F*A[K,J], A[I,K] <- -F*P^-1  (WMMA rank-16 update)
__global__ void k_invert(float* __restrict__ PL) {
  extern __shared__ float A[];  // 256 rows x STR words
  float* M = PL + (size_t)blockIdx.x * ID * ID;
  const int tid = threadIdx.x;       // 256 threads = 8 wave32
  const int lane = tid & 31;
  const int wv = tid >> 5;
  const int m = lane & 15;
  const int bk = (lane >> 4) << 1;
  const int rb = (lane >> 4) << 3;
  const int pi = tid >> 4, pj = tid & 15;  // phase-A element ownership

  // global (row-major 256) -> LDS (stride STR)
  for (int idx = tid; idx < ID * (ID / 4); idx += 256) {
    int row = idx >> 6, c4 = (idx & 63) << 2;
    *(v4f*)&A[row * STR + c4] = *(const v4f*)&M[(size_t)row * ID + c4];
  }
  __syncthreads();

  for (int kb = 0; kb < 16; ++kb) {
    const int r = kb << 4;
    // ---- Phase A: invert pivot block in place ----
    for (int k = 0; k < 16; ++k) {
      float pr  = 1.0f / A[(r + k) * STR + r + k];
      float cur = A[(r + pi) * STR + r + pj];
      float f   = A[(r + pi) * STR + r + k];
      float rkj = A[(r + k) * STR + r + pj];
      __syncthreads();
      float nv;
      if (pi == k) nv = (pj == k) ? pr : rkj * pr;
      else         nv = (pj == k) ? (-f * pr) : fmaf(-(f * pr), rkj, cur);
      A[(r + pi) * STR + r + pj] = nv;
      __syncthreads();
    }
    // ---- Phase B: scale pivot row-block ----
    for (int jj = wv; jj < 15; jj += 8) {
      int jc = ((jj < kb) ? jj : jj + 1) << 4;
      v2f af[4], bf[4];
      #pragma unroll
      for (int t = 0; t < 4; ++t) {
        int kk = (t << 2) + bk;
        af[t].x = A[(r + m) * STR + r + kk];      // P^-1 fragment
        af[t].y = A[(r + m) * STR + r + kk + 1];
        bf[t].x = A[(r + kk) * STR + jc + m];     // old A[K,J] fragment
        bf[t].y = A[(r + kk + 1) * STR + jc + m];
      }
      v8f c = {};
      #pragma unroll
      for (int t = 0; t < 4; ++t) c = wmma_k4(af[t], bf[t], c);
      #pragma unroll
      for (int v = 0; v < 8; ++v) A[(r + rb + v) * STR + jc + m] = c[v];
    }
    __syncthreads();
    // ---- Phase C: trailing rank-16 update (each wave owns whole row-blocks)
    for (int ii = wv; ii < 15; ii += 8) {
      int ir = ((ii < kb) ? ii : ii + 1) << 4;
      v2f nf[4];  // -F fragment (f32 WMMA has no neg_a modifier)
      #pragma unroll
      for (int t = 0; t < 4; ++t) {
        int kk = (t << 2) + bk;
        nf[t].x = -A[(ir + m) * STR + r + kk];
        nf[t].y = -A[(ir + m) * STR + r + kk + 1];
      }
      for (int jb = 0; jb < 16; ++jb) {
        int jc = jb << 4;
        v8f c = {};
        if (jb != kb) {
          #pragma unroll
          for (int v = 0; v < 8; ++v) c[v] = A[(ir + rb + v) * STR + jc + m];
        }
        v2f bf[4];
        #pragma unroll
        for (int t = 0; t < 4; ++t) {
          int kk = (t << 2) + bk;
          bf[t].x = A[(r + kk) * STR + jc + m];
          bf[t].y = A[(r + kk + 1) * STR + jc + m];
        }
        #pragma unroll
        for (int t = 0; t < 4; ++t) c = wmma_k4(nf[t], bf[t], c);
        #pragma unroll
        for (int v = 0; v < 8; ++v) A[(ir + rb + v) * STR + jc + m] = c[v];
      }
    }
    __syncthreads();
  }

  // LDS -> global writeback
  for (int idx = tid; idx < ID * (ID / 4); idx += 256) {
    int row = idx >> 6, c4 = (idx & 63) << 2;
    *(v4f*)&M[(size_t)row * ID + c4] = *(v4f*)&A[row * STR + c4];
  }
}

// PK[b] = Linv[b] @ RHS[b]   -> [B,256,64]
__global__ void k_pk(const float* __restrict__ Linv, const float* __restrict__ RHS,
                     float* __restrict__ PK) {
  int lane = threadIdx.x & 31;
  int tile = blockIdx.x * 4 + (threadIdx.x >> 5);  // 0..63
  int b = blockIdx.y;
  int it = (tile >> 2) << 4;
  int jt = (tile & 3) << 4;
  const float* Li = Linv + (size_t)b * ID * ID;
  const float* R = RHS + (size_t)b * ID * OD;
  int m = lane & 15;
  int bk = (lane >> 4) << 1;
  v8f c = {};
  for (int k0 = 0; k0 < ID; k0 += 4) {
    v2f a = *reinterpret_cast<const v2f*>(Li + (size_t)(it + m) * ID + k0 + bk);
    v2f bb;
    bb.x = R[(size_t)(k0 + bk + 0) * OD + jt + m];
    bb.y = R[(size_t)(k0 + bk + 1) * OD + jt + m];
    c = wmma_k4(a, bb, c);
  }
  float* out = PK + (size_t)b * ID * OD;
  int rb = (lane >> 4) << 3;
  #pragma unroll
  for (int v = 0; v < 8; ++v)
    out[(size_t)(it + rb + v) * OD + jt + m] = c[v];
}

// mu[b] = PhiQ[b] @ PK[b]   -> d_out [B,512,64]
__global__ void k_mu(const float* __restrict__ phiQ, const float* __restrict__ PK,
                     float* __restrict__ mu) {
  int lane = threadIdx.x & 31;
  int tile = blockIdx.x * 4 + (threadIdx.x >> 5);  // 0..127
  int b = blockIdx.y;
  int qt = (tile >> 2) << 4;
  int jt = (tile & 3) << 4;
  const float* Pq = phiQ + (size_t)b * QN * ID;
  const float* P = PK + (size_t)b * ID * OD;
  int m = lane & 15;
  int bk = (lane >> 4) << 1;
  v8f c = {};
  for (int k0 = 0; k0 < ID; k0 += 4) {
    v2f a = *reinterpret_cast<const v2f*>(Pq + (size_t)(qt + m) * ID + k0 + bk);
    v2f bb;
    bb.x = P[(size_t)(k0 + bk + 0) * OD + jt + m];
    bb.y = P[(size_t)(k0 + bk + 1) * OD + jt + m];
    c = wmma_k4(a, bb, c);
  }
  float* out = mu + (size_t)b * QN * OD;
  int rb = (lane >> 4) << 3;
  #pragma unroll
  for (int v = 0; v < 8; ++v)
    out[(size_t)(qt + rb + v) * OD + jt + m] = c[v];
}

// spread[b][q] += sum_j (PhiQ @ Linv)[q][j] * PhiQ[q][j]  (fused, no temp)
__global__ void k_spread(const float* __restrict__ phiQ, const float* __restrict__ Linv,
                         float* __restrict__ spread) {
  int lane = threadIdx.x & 31;
  int tile = blockIdx.x * 4 + (threadIdx.x >> 5);  // 0..511
  int b = blockIdx.y;
  int qt = (tile >> 4) << 4;
  int jt = (tile & 15) << 4;
  const float* Pq = phiQ + (size_t)b * QN * ID;
  const float* Li = Linv + (size_t)b * ID * ID;
  int m = lane & 15;
  int bk = (lane >> 4) << 1;
  v8f c = {};
  for (int k0 = 0; k0 < ID; k0 += 4) {
    v2f a = *reinterpret_cast<const v2f*>(Pq + (size_t)(qt + m) * ID + k0 + bk);
    v2f bb;
    bb.x = Li[(size_t)(k0 + bk + 0) * ID + jt + m];
    bb.y = Li[(size_t)(k0 + bk + 1) * ID + jt + m];
    c = wmma_k4(a, bb, c);
  }
  int rb = (lane >> 4) << 3;
  #pragma unroll
  for (int v = 0; v < 8; ++v) {
    int q = qt + rb + v;
    float t = c[v] * Pq[(size_t)q * ID + jt + m];
    // reduce over the 16 lanes of this half-wave (xor<16 stays in the half)
    t += __shfl_xor(t, 1, 32);
    t += __shfl_xor(t, 2, 32);
    t += __shfl_xor(t, 4, 32);
    t += __shfl_xor(t, 8, 32);
    if (m == 0) atomicAdd(&spread[(size_t)b * QN + q], t);
  }
}

// sig_pred[b,q,o,p] = (o==p) ? spread*eps : 0 -- 268 MB streaming store.
// Larger than the 192 MB L2: use non-temporal b128 stores to avoid thrashing.
__global__ void k_sig(const float* __restrict__ spread, float* __restrict__ sig) {
  int t = blockIdx.x * blockDim.x + threadIdx.x;  // over B*Q*O rows
  if (t >= BN * QN * OD) return;
  int o = t & (OD - 1);
  int bq = t >> 6;
  float s = spread[bq] * SIG_EPS;
  v4f* out = (v4f*)(sig + (size_t)t * OD);
  int oi = o >> 2, oc = o & 3;
  #pragma unroll
  for (int i = 0; i < OD / 4; ++i) {
    v4f v = {0.f, 0.f, 0.f, 0.f};
    if (i == oi) v[oc] = s;
    __builtin_nontemporal_store(v, out + i);
  }
}

// predictive_nll accumulation
__global__ void k_nll(const float* __restrict__ yq, const float* __restrict__ mu,
                      const float* __restrict__ spread, float* __restrict__ nll) {
  int t = blockIdx.x * blockDim.x + threadIdx.x;
  if (t >= BN * QN) return;
  const float* y = yq + (size_t)t * OD;
  const float* m = mu + (size_t)t * OD;
  float ss = 0.0f;
  for (int o = 0; o < OD; ++o) {
    float d = y[o] - m[o];
    ss = fmaf(d, d, ss);
  }
  float sp = spread[t];
  float quad = ss / (sp * SIG_EPS);
  float ld = (float)OD * (logf(sp) + logf(SIG_EPS));
  atomicAdd(nll, (ld + quad) * (1.0f / (float)(BN * QN)));
}

// ---------------- launcher ----------------

extern "C" void kernel_launch(void* const* d_in, const int* in_sizes, int n_in,
                              void* d_out, int out_size, void* d_ws, size_t ws_size,
                              hipStream_t stream) {
  const float* phiS = (const float*)d_in[0];
  const float* yS   = (const float*)d_in[1];
  const float* phiQ = (const float*)d_in[2];
  const float* yQ   = (const float*)d_in[3];
  const float* Kp   = (const float*)d_in[4];
  const float* La   = (const float*)d_in[5];

  float* out = (float*)d_out;
  float* mu  = out;                                        // [B,Q,O]
  float* sig = out + (size_t)BN * QN * OD;                 // [B,Q,O,O]
  float* nll = sig + (size_t)BN * QN * OD * OD;            // scalar

  // workspace layout (~13 MB, fully rewritten every call)
  float* ws = (float*)d_ws;
  float* L      = ws;                                      // 256*256
  float* LK     = L + ID * ID;                             // 256*64
  float* PL     = LK + ID * OD;                            // B*256*256 (then inverse)
  float* RHS    = PL + (size_t)BN * ID * ID;               // B*256*64
  float* PK     = RHS + (size_t)BN * ID * OD;              // B*256*64
  float* spread = PK + (size_t)BN * ID * OD;               // B*Q

  // allow 266 KB dynamic LDS for the inverse kernel (CDNA5: 320 KB/WGP)
  (void)hipFuncSetAttribute(reinterpret_cast<const void*>(k_invert),
                            hipFuncAttributeMaxDynamicSharedMemorySize,
                            ID * STR * (int)sizeof(float));

  k_init<<<(BN * QN + 255) / 256, 256, 0, stream>>>(spread, nll);
  k_L<<<(ID * ID + 255) / 256, 256, 0, stream>>>(La, L);
  k_LK<<<(ID * OD + 255) / 256, 256, 0, stream>>>(L, Kp, LK);
  k_postL<<<dim3(64, BN), 128, 0, stream>>>(phiS, L, PL);
  k_rhs<<<dim3(16, BN), 128, 0, stream>>>(phiS, yS, LK, RHS);
  k_invert<<<BN, 256, ID * STR * sizeof(float), stream>>>(PL);
  k_pk<<<dim3(16, BN), 128, 0, stream>>>(PL, RHS, PK);
  k_mu<<<dim3(32, BN), 128, 0, stream>>>(phiQ, PK, mu);
  k_spread<<<dim3(128, BN), 128, 0, stream>>>(phiQ, PL, spread);
  k_sig<<<(BN * QN * OD + 255) / 256, 256, 0, stream>>>(spread, sig);
  k_nll<<<(BN * QN + 255) / 256, 256, 0, stream>>>(yQ, mu, spread, nll);
}